// FP32ToFP64Converter_76312978916101
// MI455X (gfx1250) — compile-verified
//
#include <hip/hip_runtime.h>
#include <hip/hip_bf16.h>

// FP32 bit-vector -> FP64 bit-vector converter (boolean network collapsed to
// integer bit manipulation). Pure streaming kernel: 256 MB in, 512 MB out,
// HBM-bound (~33 us floor at 23.3 TB/s on MI455X). One thread per row.

// Native clang vector types: required by __builtin_nontemporal_load/store
// (HIP's uint4/float4 are structs and are rejected by the builtin).
typedef __attribute__((ext_vector_type(4))) unsigned int v4u;
typedef __attribute__((ext_vector_type(4))) float        v4f;

__global__ __launch_bounds__(256)
void FP32ToFP64Converter_kernel(const float* __restrict__ in,
                                float* __restrict__ out,
                                int rows) {
    const int row = blockIdx.x * blockDim.x + threadIdx.x;
    if (row >= rows) return;

    // ---- Load 32 input "bit" floats as 8 x b128 (non-temporal: read-once) ----
    const v4u* src = reinterpret_cast<const v4u*>(in) + (size_t)row * 8;
    uint32_t w[32];
    #pragma unroll
    for (int i = 0; i < 8; ++i) {
        v4u v = __builtin_nontemporal_load(&src[i]);
        w[i * 4 + 0] = v.x;
        w[i * 4 + 1] = v.y;
        w[i * 4 + 2] = v.z;
        w[i * 4 + 3] = v.w;
    }

    // ---- Pack bits. Each element is exactly 0.0f (0x00000000) or 1.0f
    //      (0x3F800000); bit = (word >> 29) & 1 (one v_bfe per element). ----
    // Layout: elem 0 = sign, elems 1..8 = exponent MSB-first,
    //         elems 9..31 = mantissa MSB-first.
    const uint32_t s_bit = (w[0] >> 29) & 1u;
    uint32_t e = 0u;
    #pragma unroll
    for (int j = 1; j <= 8; ++j)  e = (e << 1) | ((w[j] >> 29) & 1u);
    uint32_t m = 0u;
    #pragma unroll
    for (int j = 9; j <= 31; ++j) m = (m << 1) | ((w[j] >> 29) & 1u);

    // ---- FP64 pattern construction (exact integer logic) ----
    // normal/denorm-as-zero: exp = (e==0) ? 0 : e+896 ; mant = m << 29
    // e==255: Inf -> exp=0x7FF, mant=0 ; NaN -> exp=0x7FF, mant=1<<51
    uint32_t exp11 = (e == 0u) ? 0u : (e + 896u);
    uint32_t mant_hi20 = m >> 3;            // mantissa bits 51..32
    uint32_t mant_lo32 = m << 29;           // mantissa bits 31..0
    if (e == 255u) {
        exp11 = 0x7FFu;
        mant_hi20 = (m != 0u) ? 0x80000u : 0u;   // NaN: mantissa MSB only
        mant_lo32 = 0u;
    }
    const uint32_t hi = (s_bit << 31) | (exp11 << 20) | mant_hi20;
    const uint32_t lo = mant_lo32;

    // ---- Unpack 64 bits MSB-first into 64 floats, store as 16 x b128 NT ----
    v4f* dst = reinterpret_cast<v4f*>(out) + (size_t)row * 16;
    #pragma unroll
    for (int q = 0; q < 16; ++q) {
        const uint32_t word = (q < 8) ? hi : lo;
        const int base = (q * 4) & 31;      // 0,4,8,...,28 within the word
        v4f t;
        t.x = (float)((word >> (31 - base)) & 1u);  // v_bfe + v_cvt_f32_u32
        t.y = (float)((word >> (30 - base)) & 1u);
        t.z = (float)((word >> (29 - base)) & 1u);
        t.w = (float)((word >> (28 - base)) & 1u);
        __builtin_nontemporal_store(t, &dst[q]);
    }
}

extern "C" void kernel_launch(void* const* d_in, const int* in_sizes, int n_in,
                              void* d_out, int out_size, void* d_ws, size_t ws_size,
                              hipStream_t stream) {
    const float* in  = (const float*)d_in[0];
    float*       out = (float*)d_out;
    const int rows = in_sizes[0] / 32;          // B = 2^21
    const int threads = 256;                    // 8 wave32 waves per block
    const int blocks = (rows + threads - 1) / threads;
    FP32ToFP64Converter_kernel<<<blocks, threads, 0, stream>>>(in, out, rows);
}